// Model_23880018165857
// MI455X (gfx1250) — compile-verified
//
#include <hip/hip_runtime.h>
#include <stdint.h>

// Problem constants (from reference): points (32, 200000, 3) f32 -> indices (32, 128) i64
#define NB        32
#define NPTS      200000
#define MS        128
#define WGS       512                    // 16 wave32 per workgroup
#define CHUNKS    8                      // workgroups cooperating per batch
#define CHUNK_N   25000                  // NPTS / CHUNKS
#define NSLICE    49                     // ceil(CHUNK_N / WGS); last slice partial (424 pts)
#define CHUNK_BYTES (CHUNK_N * 12)       // 300000 bytes of xyz per chunk
#define NWAVE     (WGS / 32)

// Dynamic LDS layout (bytes)
#define LMIN_OFF   0
#define PBUF_OFF   (CHUNK_N * 4)                  // 100000 : per-wave double-buffer point tiles
#define PBUF_DW    (NWAVE * 2 * 96)               // 16 waves * 2 bufs * 96 dwords (32 pts * xyz)
#define WRED_OFF   (PBUF_OFF + PBUF_DW * 4)       // 112288 (8B aligned)
#define BC_OFF     (WRED_OFF + NWAVE * 8)         // 112416 : broadcast {idx, cx, cy, cz}
#define SMEM_BYTES (BC_OFF + 16)                  // 112432

// CDNA5 async global->LDS copy: copies 4B per lane (128B per wave32 instr).
#define ASYNC_TO_LDS_B32(ldsoff, goff, sbase)                            \
  asm volatile("global_load_async_to_lds_b32 %0, %1, %2"                 \
               :: "v"(ldsoff), "v"(goff), "s"(sbase) : "memory")

extern "C" __global__ __launch_bounds__(WGS, 1)
void fps_kernel(const float* __restrict__ points,
                long long* __restrict__ out,
                unsigned long long* __restrict__ best,
                unsigned int* __restrict__ cnt) {
  extern __shared__ char smem[];
  float*              lmin = (float*)(smem + LMIN_OFF);
  uint32_t*           pbuf = (uint32_t*)(smem + PBUF_OFF);
  unsigned long long* wred = (unsigned long long*)(smem + WRED_OFF);
  uint32_t*           bc   = (uint32_t*)(smem + BC_OFF);

  const int t    = threadIdx.x;
  const int lane = t & 31;
  const int wid  = t >> 5;
  const int b     = blockIdx.x >> 3;
  const int chunk = blockIdx.x & 7;
  const uint32_t chunkStart = (uint32_t)chunk * CHUNK_N;
  const float* __restrict__ pbatch = points + (size_t)b * NPTS * 3;   // batch base
  const float* __restrict__ cbase  = pbatch + (size_t)chunkStart * 3; // chunk base (SGPR pair)

  // init min-dist (LDS-resident across all rounds)
  for (int l = t; l < CHUNK_N; l += WGS) lmin[l] = __builtin_inff();

  if (t == 0) {
    bc[0] = 0u;                              // initial farthest index = 0
    bc[1] = __float_as_uint(pbatch[0]);
    bc[2] = __float_as_uint(pbatch[1]);
    bc[3] = __float_as_uint(pbatch[2]);
    if (chunk == 0) out[(size_t)b * MS] = 0; // chosen[0] == 0 per reference scan
  }
  __syncthreads();

  // Per-wave LDS staging bases (generic ptr low 32 bits == LDS byte offset)
  const uint32_t ldsbase =
      (uint32_t)(uintptr_t)pbuf + (uint32_t)(wid * 2 * 96 * 4) + (uint32_t)(lane * 4);
  const uint32_t* __restrict__ pb0 = pbuf + (wid * 2 + 0) * 96 + 3 * lane;
  const uint32_t* __restrict__ pb1 = pbuf + (wid * 2 + 1) * 96 + 3 * lane;
  const uint32_t glim = CHUNK_BYTES - 4;     // clamp target for tail slice
  const uint32_t gbase0 = (uint32_t)(wid * 32) * 12u + (uint32_t)(lane * 4); // slice-0 offsets

  // Prefetch slice 0 into parity-0 buffer (3 x 128B per wave)
  ASYNC_TO_LDS_B32(ldsbase,        gbase0,        cbase);
  ASYNC_TO_LDS_B32(ldsbase + 128u, gbase0 + 128u, cbase);
  ASYNC_TO_LDS_B32(ldsbase + 256u, gbase0 + 256u, cbase);

  for (int it = 0; it < MS - 1; ++it) {
    const uint32_t clocal = bc[0] - chunkStart;  // uniform; wraps huge if not our chunk
    const float    cx = __uint_as_float(bc[1]);
    const float    cy = __uint_as_float(bc[2]);
    const float    cz = __uint_as_float(bc[3]);

    float    bestm = -2.0f;   // below the -1 sentinel
    uint32_t bestl = 0;

    #pragma unroll 2
    for (int j = 0; j < NSLICE; ++j) {
      const int par = j & 1;
      if (j + 1 < NSLICE) {
        // software-pipelined async prefetch of slice j+1 into the other buffer
        uint32_t go = (uint32_t)((j + 1) * WGS) * 12u + gbase0;
        uint32_t a0 = go;         if (a0 > glim) a0 = glim;   // tail clamp (slice 48)
        uint32_t a1 = go + 128u;  if (a1 > glim) a1 = glim;
        uint32_t a2 = go + 256u;  if (a2 > glim) a2 = glim;
        uint32_t l0 = ldsbase + (uint32_t)((par ^ 1) * 384);
        ASYNC_TO_LDS_B32(l0,        a0, cbase);
        ASYNC_TO_LDS_B32(l0 + 128u, a1, cbase);
        ASYNC_TO_LDS_B32(l0 + 256u, a2, cbase);
        asm volatile("s_wait_asynccnt 0x3" ::: "memory"); // slice j complete
      } else {
        asm volatile("s_wait_asynccnt 0x0" ::: "memory"); // drain last slice
      }

      const uint32_t l = (uint32_t)(j * WGS + t);
      if (l < CHUNK_N) {
        const uint32_t* pb = par ? pb1 : pb0;
        float x = __uint_as_float(pb[0]);
        float y = __uint_as_float(pb[1]);
        float z = __uint_as_float(pb[2]);
        float dx = x - cx, dy = y - cy, dz = z - cz;
        float d  = fmaf(dz, dz, fmaf(dy, dy, dx * dx));
        float m  = fminf(lmin[l], d);
        if (l == clocal) m = -1.0f;                 // exclude current center
        lmin[l] = m;
        if (m > bestm) { bestm = m; bestl = l; }    // strict > keeps lowest l per thread
      }
    }

    // Prefetch slice 0 for the NEXT round now; it completes under the barrier/spin below.
    ASYNC_TO_LDS_B32(ldsbase,        gbase0,        cbase);
    ASYNC_TO_LDS_B32(ldsbase + 128u, gbase0 + 128u, cbase);
    ASYNC_TO_LDS_B32(ldsbase + 256u, gbase0 + 256u, cbase);

    // argmax key: value bits of (m + 1.0) (maps -1 sentinel -> 0, monotone for m >= -1),
    // low 32 = ~global_index so ties pick the LOWEST index (argmax first-occurrence).
    unsigned long long key =
        ((unsigned long long)__float_as_uint(bestm + 1.0f) << 32) |
        (unsigned long long)(~(chunkStart + bestl));

    for (int o = 16; o; o >>= 1) {
      unsigned long long oth = __shfl_xor(key, o, 32);
      if (oth > key) key = oth;
    }
    if (lane == 0) wred[wid] = key;
    __syncthreads();

    if (wid == 0) {
      unsigned long long k2 = (lane < NWAVE) ? wred[lane] : 0ull;
      for (int o = 8; o; o >>= 1) {
        unsigned long long oth = __shfl_xor(k2, o, 32);
        if (oth > k2) k2 = oth;
      }
      if (lane == 0) {
        const uint32_t slot = (uint32_t)b * MS + (uint32_t)it;
        atomicMax(&best[slot], k2);
        // release arrival: orders the atomicMax before the counter bump
        __hip_atomic_fetch_add(&cnt[slot], 1u, __ATOMIC_RELEASE, __HIP_MEMORY_SCOPE_AGENT);
        while (__hip_atomic_load(&cnt[slot], __ATOMIC_ACQUIRE, __HIP_MEMORY_SCOPE_AGENT) < CHUNKS)
          __builtin_amdgcn_s_sleep(2);
        unsigned long long fin =
            __hip_atomic_load(&best[slot], __ATOMIC_ACQUIRE, __HIP_MEMORY_SCOPE_AGENT);
        uint32_t cn = ~(uint32_t)fin;          // batch-local winning index
        bc[0] = cn;
        const float* cp = pbatch + (size_t)cn * 3;
        bc[1] = __float_as_uint(cp[0]);
        bc[2] = __float_as_uint(cp[1]);
        bc[3] = __float_as_uint(cp[2]);
        if (chunk == 0) out[(size_t)b * MS + it + 1] = (long long)cn;
      }
    }
    __syncthreads();
  }
}

extern "C" __global__ void fps_ws_init(unsigned long long* __restrict__ best,
                                       unsigned int* __restrict__ cnt) {
  int i = blockIdx.x * blockDim.x + threadIdx.x;
  if (i < NB * MS) { best[i] = 0ull; cnt[i] = 0u; }
}

extern "C" void kernel_launch(void* const* d_in, const int* in_sizes, int n_in,
                              void* d_out, int out_size, void* d_ws, size_t ws_size,
                              hipStream_t stream) {
  const float* points = (const float*)d_in[0];
  long long* out = (long long*)d_out;

  unsigned long long* best = (unsigned long long*)d_ws;                 // NB*MS u64
  unsigned int* cnt = (unsigned int*)((char*)d_ws + NB * MS * sizeof(unsigned long long));

  (void)in_sizes; (void)n_in; (void)out_size; (void)ws_size;

  // Allow >64KB dynamic LDS (deterministic host-side config; not a stream op).
  hipFuncSetAttribute((const void*)fps_kernel,
                      hipFuncAttributeMaxDynamicSharedMemorySize, SMEM_BYTES);

  fps_ws_init<<<(NB * MS + 255) / 256, 256, 0, stream>>>(best, cnt);
  fps_kernel<<<NB * CHUNKS, WGS, SMEM_BYTES, stream>>>(points, out, best, cnt);
}